// DCP_6889127542959
// MI455X (gfx1250) — compile-verified
//
#include <hip/hip_runtime.h>
#include <stdint.h>

// Problem constants (reference: x of shape (16, 3, 640, 640) fp32)
#define B    16
#define C    3
#define H    640
#define W    640
#define P    (H * W)          // 409600 pixels per image
#define KSEL 409              // int(H*W*0.001)
#define RAD  7                // 15x15 window radius
#define NBINS 4096
#define CAP  65536            // candidate cap per batch

#define INF __builtin_inff()

// ---------------------------------------------------------------------------
// Async global->LDS staging via CDNA5 inline asm (GLOBAL_LOAD_ASYNC_TO_LDS_B32,
// tracked by ASYNCcnt; CDNA5 ISA).
// The clang builtin uses cuda_device/cuda_shared address-space params that HIP
// source cannot spell, so we go through the assembler directly.
// ---------------------------------------------------------------------------
#define USE_ASYNC_LDS 1

__device__ __forceinline__ void async_copy_f32(float* lds_dst, const float* gsrc) {
#if USE_ASYNC_LDS
  // Low 32 bits of a generic LDS pointer == LDS byte offset.
  unsigned lds_off = (unsigned)(uintptr_t)lds_dst;
  asm volatile("global_load_async_to_lds_b32 %0, %1, off"
               :
               : "v"(lds_off), "v"(gsrc)
               : "memory");
#else
  *lds_dst = *gsrc;
#endif
}

__device__ __forceinline__ void wait_async_lds() {
#if USE_ASYNC_LDS
  asm volatile("s_wait_asynccnt 0" ::: "memory");
#endif
}

// ---------------------------------------------------------------------------
// K1/K9: fused per-pixel channel-min (with per-(b,c) positive scale) +
//        horizontal 15-tap min, +inf out-of-range padding.
// grid (3, 640, 16), block 256
// ---------------------------------------------------------------------------
__global__ void cmin_hmin_kernel(const float* __restrict__ x,
                                 const float* __restrict__ scale,
                                 float* __restrict__ out) {
  __shared__ float smem[4 * 272];  // ch0, ch1, ch2, cmin
  float* ch0 = smem;
  float* ch1 = smem + 272;
  float* ch2 = smem + 544;
  float* cm  = smem + 816;

  const int tid = threadIdx.x;
  const int x0  = blockIdx.x * 256;
  const int y   = blockIdx.y;
  const int b   = blockIdx.z;

  const float s0 = scale[b * 3 + 0];
  const float s1 = scale[b * 3 + 1];
  const float s2 = scale[b * 3 + 2];

  const float* base = x + ((size_t)b * C) * (size_t)P + (size_t)y * W;

  // Stage 270 slots (x0-7 .. x0+262) of all 3 channels into LDS.
  for (int j = tid; j < 256 + 2 * RAD; j += 256) {
    int gx = x0 - RAD + j;
    if (gx >= 0 && gx < W) {
      async_copy_f32(&ch0[j], base + gx);
      async_copy_f32(&ch1[j], base + P + gx);
      async_copy_f32(&ch2[j], base + 2 * P + gx);
    }
  }
  wait_async_lds();
  __syncthreads();

  for (int j = tid; j < 256 + 2 * RAD; j += 256) {
    int gx = x0 - RAD + j;
    cm[j] = (gx >= 0 && gx < W)
                ? fminf(fminf(ch0[j] * s0, ch1[j] * s1), ch2[j] * s2)
                : INF;
  }
  __syncthreads();

  const int xo = x0 + tid;
  if (xo < W) {
    float m = cm[tid];
#pragma unroll
    for (int k = 1; k < 15; ++k) m = fminf(m, cm[tid + k]);
    out[(size_t)b * P + (size_t)y * W + xo] = m;
  }
}

// ---------------------------------------------------------------------------
// K2: vertical 15-tap min (reads hit L2; whole 26MB intermediate is resident)
// grid (3, 640, 16), block 256
// ---------------------------------------------------------------------------
__global__ void vmin_kernel(const float* __restrict__ in, float* __restrict__ out) {
  const int xo = blockIdx.x * 256 + threadIdx.x;
  const int y  = blockIdx.y;
  const int b  = blockIdx.z;
  if (xo >= W) return;
  const float* col = in + (size_t)b * P + xo;
  float m = INF;
#pragma unroll
  for (int dy = -RAD; dy <= RAD; ++dy) {
    int yy = y + dy;
    if ((unsigned)yy < (unsigned)H) m = fminf(m, col[(size_t)yy * W]);
  }
  out[(size_t)b * P + (size_t)y * W + xo] = m;
}

// ---------------------------------------------------------------------------
// K3: per-batch 4096-bin histogram of dark-value bit patterns (bits >> 19).
// dark in [0,1] => bits <= 0x3F800000, bin <= 2032 < 4096.
// grid (7, 16), block 1024; LDS histogram then global atomic merge.
// ---------------------------------------------------------------------------
__global__ void hist_kernel(const float* __restrict__ dark, unsigned* __restrict__ hist) {
  __shared__ unsigned h[NBINS];
  const int b = blockIdx.y;
  for (int i = threadIdx.x; i < NBINS; i += 1024) h[i] = 0u;
  __syncthreads();
  const float* d = dark + (size_t)b * P;
  const int base = blockIdx.x * 65536;
  for (int k = 0; k < 64; ++k) {
    int i = base + k * 1024 + threadIdx.x;
    if (i < P) {
      unsigned bits = __float_as_uint(d[i]);
      atomicAdd(&h[bits >> 19], 1u);
    }
  }
  __syncthreads();
  for (int i = threadIdx.x; i < NBINS; i += 1024) {
    unsigned v = h[i];
    if (v) atomicAdd(&hist[b * NBINS + i], v);
  }
}

// ---------------------------------------------------------------------------
// K4: scan histogram from the top: find the bin containing rank KSEL and the
// residual rank within it.  state[b*2+0]=bin, state[b*2+1]=remaining (1-based).
// ---------------------------------------------------------------------------
__global__ void scan_kernel(const unsigned* __restrict__ hist, int* __restrict__ state) {
  const int b = threadIdx.x;
  if (b >= B) return;
  const unsigned* h = hist + b * NBINS;
  int cum = 0;
  for (int bin = NBINS - 1; bin >= 0; --bin) {
    int c = (int)h[bin];
    if (cum + c >= KSEL) {
      state[b * 2 + 0] = bin;
      state[b * 2 + 1] = KSEL - cum;  // remaining rank inside this bin
      return;
    }
    cum += c;
  }
}

// ---------------------------------------------------------------------------
// K5: compact candidates of the cutoff bin as 64-bit keys.
// key = (value_bits << 32) | (0xFFFFFFFF - idx)  -> larger key == larger value,
// ties broken toward smaller index (matches lax.top_k tie-breaking).
// grid (1600, 16), block 256
// ---------------------------------------------------------------------------
__global__ void gather_kernel(const float* __restrict__ dark, const int* __restrict__ state,
                              unsigned long long* __restrict__ cand, unsigned* __restrict__ cnt) {
  const int b = blockIdx.y;
  const int i = blockIdx.x * 256 + threadIdx.x;  // 0..P-1
  unsigned bits = __float_as_uint(dark[(size_t)b * P + i]);
  if ((int)(bits >> 19) == state[b * 2 + 0]) {
    unsigned pos = atomicAdd(&cnt[b], 1u);
    if (pos < CAP)
      cand[(size_t)b * CAP + pos] =
          ((unsigned long long)bits << 32) | (0xFFFFFFFFu - (unsigned)i);
  }
}

// ---------------------------------------------------------------------------
// K6: per-batch MSB-first byte-radix select of the `remaining`-th largest key
// among the candidates -> kstar[b] (the exact rank-KSEL key overall).
// grid (16), block 256
// ---------------------------------------------------------------------------
__global__ void select_kernel(const unsigned long long* __restrict__ cand,
                              const unsigned* __restrict__ cnt,
                              const int* __restrict__ state,
                              unsigned long long* __restrict__ kstar) {
  __shared__ unsigned h[256];
  __shared__ unsigned long long sprefix;
  __shared__ int sr;
  const int b = blockIdx.x;
  const unsigned long long* c = cand + (size_t)b * CAP;
  const int n = (int)min(cnt[b], (unsigned)CAP);
  if (threadIdx.x == 0) {
    sprefix = 0ull;
    int r = state[b * 2 + 1];
    sr = (r < 1) ? 1 : ((r > n) ? n : r);
  }
  __syncthreads();
  for (int byte = 7; byte >= 0; --byte) {
    h[threadIdx.x] = 0u;
    __syncthreads();
    unsigned long long prefix = sprefix;
    for (int i = threadIdx.x; i < n; i += 256) {
      unsigned long long k = c[i];
      unsigned long long hi = (byte < 7) ? (k >> (8 * (byte + 1))) : 0ull;
      if (hi == prefix) atomicAdd(&h[(unsigned)((k >> (8 * byte)) & 0xFF)], 1u);
    }
    __syncthreads();
    if (threadIdx.x == 0) {
      int r = sr, cum = 0, sel = 0;
      for (int bin = 255; bin >= 0; --bin) {
        int cc = (int)h[bin];
        if (cum + cc >= r) { sel = bin; sr = r - cum; break; }
        cum += cc;
      }
      sprefix = (prefix << 8) | (unsigned)sel;
    }
    __syncthreads();
  }
  if (threadIdx.x == 0) kstar[b] = sprefix;
}

// ---------------------------------------------------------------------------
// K7: atmosphere: per-(b,c) max of x_norm over the exactly-KSEL selected
// pixels (key >= kstar).  Nonnegative floats -> uint atomicMax on bits.
// grid (1600, 16), block 256
// ---------------------------------------------------------------------------
__global__ void atmos_kernel(const float* __restrict__ dark, const float* __restrict__ x,
                             const unsigned long long* __restrict__ kstar,
                             unsigned* __restrict__ atmosU) {
  const int b = blockIdx.y;
  const int i = blockIdx.x * 256 + threadIdx.x;
  unsigned bits = __float_as_uint(dark[(size_t)b * P + i]);
  unsigned long long key = ((unsigned long long)bits << 32) | (0xFFFFFFFFu - (unsigned)i);
  if (key >= kstar[b]) {
    const float inv255 = 1.0f / 255.0f;
#pragma unroll
    for (int ch = 0; ch < C; ++ch) {
      float v = x[((size_t)b * C + ch) * (size_t)P + i] * inv255;
      atomicMax(&atmosU[b * 3 + ch], __float_as_uint(v));
    }
  }
}

// ---------------------------------------------------------------------------
// K0/K8: per-(b,c) scales.  Pass 1: 1/255.  Pass 2: 1/(255*(A+1e-8)).
// ---------------------------------------------------------------------------
__global__ void scale_init_kernel(float* __restrict__ scale) {
  int i = threadIdx.x;
  if (i < B * C) scale[i] = 1.0f / 255.0f;
}

__global__ void scale2_kernel(const unsigned* __restrict__ atmosU, float* __restrict__ scale) {
  int i = threadIdx.x;
  if (i < B * C) {
    float a = __uint_as_float(atmosU[i]);
    scale[i] = 1.0f / (255.0f * (a + 1e-8f));
  }
}

// ---------------------------------------------------------------------------
// K10: fused vertical 15-min of the 2nd-pass hmin + recovery.
// t = max(1 - 0.95*dark2, 0.1); out = clip((x/255 - A)/t + A, 0, 1) * 255
// grid (3, 640, 16), block 256
// ---------------------------------------------------------------------------
__global__ void final_kernel(const float* __restrict__ hminA, const float* __restrict__ x,
                             const unsigned* __restrict__ atmosU, float* __restrict__ out) {
  const int xo = blockIdx.x * 256 + threadIdx.x;
  const int y  = blockIdx.y;
  const int b  = blockIdx.z;
  if (xo >= W) return;
  const float* col = hminA + (size_t)b * P + xo;
  float m = INF;
#pragma unroll
  for (int dy = -RAD; dy <= RAD; ++dy) {
    int yy = y + dy;
    if ((unsigned)yy < (unsigned)H) m = fminf(m, col[(size_t)yy * W]);
  }
  const float t = fmaxf(1.0f - 0.95f * m, 0.1f);
  const float inv255 = 1.0f / 255.0f;
  const size_t pix = (size_t)y * W + xo;
#pragma unroll
  for (int ch = 0; ch < C; ++ch) {
    float A  = __uint_as_float(atmosU[b * 3 + ch]);
    float xn = x[((size_t)b * C + ch) * (size_t)P + pix] * inv255;
    float r  = (xn - A) / t + A;
    r = fminf(fmaxf(r, 0.0f), 1.0f);
    out[((size_t)b * C + ch) * (size_t)P + pix] = r * 255.0f;
  }
}

// ---------------------------------------------------------------------------
// Workspace layout (bytes), 256-aligned; ~61 MB total.
// ---------------------------------------------------------------------------
#define OFF_HIST   ((size_t)0)                          // 16*4096*4 = 262144
#define OFF_CNT    ((size_t)262144)                     // 64
#define OFF_ATMOS  ((size_t)262208)                     // 192
#define MEMSET_BYTES ((size_t)262400)                   // hist + cnt + atmos
#define OFF_KSTAR  ((size_t)262400)                     // 128
#define OFF_STATE  ((size_t)262528)                     // 128
#define OFF_SCALE  ((size_t)262656)                     // 192
#define OFF_CAND   ((size_t)262912)                     // 16*65536*8 = 8388608
#define OFF_W1     ((size_t)8651520)                    // 26214400
#define OFF_W2     ((size_t)34865920)                   // 26214400
// end = 61080320

extern "C" void kernel_launch(void* const* d_in, const int* in_sizes, int n_in,
                              void* d_out, int out_size, void* d_ws, size_t ws_size,
                              hipStream_t stream) {
  const float* x  = (const float*)d_in[0];
  float* out      = (float*)d_out;
  char* ws        = (char*)d_ws;

  unsigned* hist             = (unsigned*)(ws + OFF_HIST);
  unsigned* cnt              = (unsigned*)(ws + OFF_CNT);
  unsigned* atmosU           = (unsigned*)(ws + OFF_ATMOS);
  unsigned long long* kstar  = (unsigned long long*)(ws + OFF_KSTAR);
  int* state                 = (int*)(ws + OFF_STATE);
  float* scale               = (float*)(ws + OFF_SCALE);
  unsigned long long* cand   = (unsigned long long*)(ws + OFF_CAND);
  float* W1                  = (float*)(ws + OFF_W1);   // hmin intermediate
  float* W2                  = (float*)(ws + OFF_W2);   // dark channel

  (void)in_sizes; (void)n_in; (void)out_size; (void)ws_size;

  // zero hist / candidate counters / atmosphere accumulators
  (void)hipMemsetAsync(ws, 0, MEMSET_BYTES, stream);

  const dim3 gridRow((W + 255) / 256, H, B);  // (3, 640, 16)

  scale_init_kernel<<<1, 64, 0, stream>>>(scale);
  cmin_hmin_kernel<<<gridRow, 256, 0, stream>>>(x, scale, W1);        // pass 1 hmin
  vmin_kernel<<<gridRow, 256, 0, stream>>>(W1, W2);                   // dark channel
  hist_kernel<<<dim3((P + 65535) / 65536, B), 1024, 0, stream>>>(W2, hist);
  scan_kernel<<<1, 32, 0, stream>>>(hist, state);
  gather_kernel<<<dim3(P / 256, B), 256, 0, stream>>>(W2, state, cand, cnt);
  select_kernel<<<B, 256, 0, stream>>>(cand, cnt, state, kstar);
  atmos_kernel<<<dim3(P / 256, B), 256, 0, stream>>>(W2, x, kstar, atmosU);
  scale2_kernel<<<1, 64, 0, stream>>>(atmosU, scale);
  cmin_hmin_kernel<<<gridRow, 256, 0, stream>>>(x, scale, W1);        // pass 2 hmin
  final_kernel<<<gridRow, 256, 0, stream>>>(W1, x, atmosU, out);      // fused vmin + recover
}